// SiameseNetwork_37185826848826
// MI455X (gfx1250) — compile-verified
//
#include <hip/hip_runtime.h>
#include <math.h>

// Problem constants (from reference): N=128 nodes, F=128 feats, H=8 heads,
// ATTR=3, M=N*N=16384 edges, HU={16,8}, HUE={4,2}.
#define NN 128
#define FF 128
#define HH 8
#define MM 16384
#define OUTW 960   // 3 attrs * 320 features per node
#define ATTRW 320

typedef __attribute__((ext_vector_type(2))) float v2f;
typedef __attribute__((ext_vector_type(8))) float v8f;

__device__ __forceinline__ float leaky02(float x) { return x >= 0.f ? x : 0.2f * x; }

// D = A(16x4,f32) * B(4x16,f32) + C   (CDNA5 V_WMMA_F32_16X16X4_F32, wave32)
__device__ __forceinline__ v8f wmma4(v2f a, v2f b, v8f c) {
  return __builtin_amdgcn_wmma_f32_16x16x4_f32(
      /*neg_a=*/false, a, /*neg_b=*/false, b,
      /*c_mod=*/(short)0, c, /*reuse_a=*/false, /*reuse_b=*/false);
}

// ---------------------------------------------------------------------------
// Edge-feature layer 0 (attr-independent, computed once):
//   featE0[m, h*4+k] = sum_f E_e2n[m,f] * We0[h,f,k],  E_e2n[m,f] = E[f,m]
__global__ void edge_feat0_kernel(const float* __restrict__ E,
                                  const float* __restrict__ We0,   // [8,3,4]
                                  const float* __restrict__ ae0n,  // [8,4]
                                  float* __restrict__ featE0,      // [M,32]
                                  float* __restrict__ aedge0) {    // [8,M]
  int m = blockIdx.x * 256 + threadIdx.x;
  if (m >= MM) return;
  float e0 = E[m], e1 = E[MM + m], e2 = E[2 * MM + m];
  for (int h = 0; h < HH; ++h) {
    float s = 0.f;
    for (int k = 0; k < 4; ++k) {
      float v = e0 * We0[h * 12 + k] + e1 * We0[h * 12 + 4 + k] + e2 * We0[h * 12 + 8 + k];
      featE0[m * 32 + h * 4 + k] = v;
      s += v * ae0n[h * 4 + k];
    }
    aedge0[h * MM + m] = s;
  }
}

// Edge-feature layer 1: featE1[m, h*2+k] = sum_c featE0[m,c] * We1[h,c,k]
__global__ void edge_feat1_kernel(const float* __restrict__ featE0,  // [M,32]
                                  const float* __restrict__ We1,     // [8,32,2]
                                  const float* __restrict__ ae1n,    // [8,2]
                                  float* __restrict__ featE1,        // [M,16]
                                  float* __restrict__ aedge1) {      // [8,M]
  int m = blockIdx.x * 256 + threadIdx.x;
  if (m >= MM) return;
  float x[32];
#pragma unroll
  for (int c = 0; c < 32; ++c) x[c] = featE0[m * 32 + c];
  for (int h = 0; h < HH; ++h) {
    float s = 0.f;
    for (int k = 0; k < 2; ++k) {
      float v = 0.f;
#pragma unroll
      for (int c = 0; c < 32; ++c) v += x[c] * We1[(h * 32 + c) * 2 + k];
      featE1[m * 16 + h * 2 + k] = v;
      s += v * ae1n[h * 2 + k];
    }
    aedge1[h * MM + m] = s;
  }
}

// ---------------------------------------------------------------------------
// Pad+transpose node weights: W[8][128][HU] -> Wt[8][16][128], zero for c>=HU.
__global__ void padt_weight_kernel(const float* __restrict__ W, int HU,
                                   float* __restrict__ Wt) {
  int idx = blockIdx.x * 256 + threadIdx.x;  // 8*16*128
  if (idx >= HH * 16 * FF) return;
  int k = idx % FF;
  int c = (idx / FF) % 16;
  int h = idx / (FF * 16);
  Wt[idx] = (c < HU) ? W[(h * FF + k) * HU + c] : 0.f;
}

// 128x128 transpose: outt[c][r] = in[r][c]
__global__ void transpose128_kernel(const float* __restrict__ in,
                                    float* __restrict__ outt) {
  int idx = blockIdx.x * 256 + threadIdx.x;
  if (idx >= NN * NN) return;
  int c = idx % NN, r = idx / NN;
  outt[c * NN + r] = in[idx];
}

// ---------------------------------------------------------------------------
// feat[h] = X_head[128,128] @ W[h][128,HU]  (WMMA f32 16x16x4, K=128)
// B comes pre-transposed+padded (Wt[8][16][128]) -> unconditional float2 loads.
// feat stored padded stride 16; padding columns come out exactly 0.
template <int HU>
__global__ void node_feat_wmma_kernel(const float* __restrict__ Xh,  // [128,128]
                                      const float* __restrict__ Wt,  // [8,16,128]
                                      float* __restrict__ feat) {    // [8,128,16]
  int mtile = blockIdx.x, h = blockIdx.y;
  int lane = threadIdx.x & 31;
  int r = lane & 15;
  int kh = (lane >> 4) << 1;
  int hi = lane >> 4;
  const float* arow = Xh + (mtile * 16 + r) * FF + kh;
  const float* brow = Wt + (h * 16 + r) * FF + kh;
  v8f c = {};
#pragma unroll 8
  for (int k0 = 0; k0 < FF; k0 += 4) {
    v2f a = *(const v2f*)(arow + k0);
    v2f b = *(const v2f*)(brow + k0);
    c = wmma4(a, b, c);
  }
#pragma unroll
  for (int v = 0; v < 8; ++v) {
    int row = mtile * 16 + v + 8 * hi;
    feat[(h * NN + row) * 16 + r] = c[v];
  }
}

// a_self[h,n] = sum_k feat[h,n,k] * ae_s[h,k]
template <int HU>
__global__ void a_self_kernel(const float* __restrict__ feat,
                              const float* __restrict__ aes,
                              float* __restrict__ a_self) {
  int idx = blockIdx.x * 256 + threadIdx.x;
  if (idx >= HH * NN) return;
  int h = idx / NN, n = idx % NN;
  float s = 0.f;
#pragma unroll
  for (int k = 0; k < HU; ++k) s += feat[(h * NN + n) * 16 + k] * aes[h * HU + k];
  a_self[idx] = s;
}

// Fused edge-to-node attention: one block per (h,n) row over M=16384 edges.
// nfe[h,n,k] = sum_m softmax_m(leaky(a_self+a_edge)+e2n)[m] * featE[m,h*KE+k] + be[h,k]
template <int KE, int RS>
__global__ void alpha_e_nfe_kernel(const float* __restrict__ a_self,  // [8,128]
                                   const float* __restrict__ a_edge,  // [8,M]
                                   const float* __restrict__ e2n,     // [128,M]
                                   const float* __restrict__ featE,   // [M,RS]
                                   const float* __restrict__ be,      // [8,KE]
                                   float* __restrict__ nfe) {         // [8,128,KE]
  int n = blockIdx.x, h = blockIdx.y, tid = threadIdx.x;
  __shared__ float red[256];
  float as = a_self[h * NN + n];
  const float* ae = a_edge + h * MM;
  const float* ad = e2n + (size_t)n * MM;
  float mx = -3.402823e38f;
  for (int m = tid; m < MM; m += 256) {
    __builtin_prefetch(ad + m + 2048, 0, 1);
    float z = leaky02(as + ae[m]) + ad[m];
    mx = fmaxf(mx, z);
  }
  red[tid] = mx; __syncthreads();
  for (int s = 128; s > 0; s >>= 1) { if (tid < s) red[tid] = fmaxf(red[tid], red[tid + s]); __syncthreads(); }
  mx = red[0]; __syncthreads();

  float sum = 0.f;
  float acc[KE];
#pragma unroll
  for (int k = 0; k < KE; ++k) acc[k] = 0.f;
  for (int m = tid; m < MM; m += 256) {
    float z = leaky02(as + ae[m]) + ad[m];
    float w = expf(z - mx);
    sum += w;
    const float* fe = featE + m * RS + h * KE;
#pragma unroll
    for (int k = 0; k < KE; ++k) acc[k] += w * fe[k];
  }
  red[tid] = sum; __syncthreads();
  for (int s = 128; s > 0; s >>= 1) { if (tid < s) red[tid] += red[tid + s]; __syncthreads(); }
  sum = red[0]; __syncthreads();
#pragma unroll
  for (int k = 0; k < KE; ++k) {
    red[tid] = acc[k]; __syncthreads();
    for (int s = 128; s > 0; s >>= 1) { if (tid < s) red[tid] += red[tid + s]; __syncthreads(); }
    if (tid == 0) nfe[(h * NN + n) * KE + k] = red[0] / sum + be[h * KE + k];
    __syncthreads();
  }
}

// feat2 = concat(feat, nfe) @ Wct + bct; written TRANSPOSED+padded:
//   feat2t[h][k][n], zero for k>=HU.  Also s2/n2 = feat2 . an_{s,n}.
template <int HU, int KE>
__global__ void feat2_kernel(const float* __restrict__ feat,  // [8,128,16]
                             const float* __restrict__ nfe,   // [8,128,KE]
                             const float* __restrict__ Wct,   // [8, HU+KE, HU]
                             const float* __restrict__ bct,   // [8, HU]
                             const float* __restrict__ ans,   // [8, HU]
                             const float* __restrict__ ann,   // [8, HU]
                             float* __restrict__ feat2t,      // [8,16,128]
                             float* __restrict__ s2, float* __restrict__ n2) {
  int idx = blockIdx.x * 256 + threadIdx.x;
  if (idx >= HH * NN) return;
  int h = idx / NN, n = idx % NN;
  const int C = HU + KE;
  float in[C];
#pragma unroll
  for (int c = 0; c < HU; ++c) in[c] = feat[(h * NN + n) * 16 + c];
#pragma unroll
  for (int c = 0; c < KE; ++c) in[HU + c] = nfe[(h * NN + n) * KE + c];
  float sa = 0.f, na = 0.f;
#pragma unroll
  for (int k = 0; k < HU; ++k) {
    float f = bct[h * HU + k];
#pragma unroll
    for (int c = 0; c < C; ++c) f += in[c] * Wct[(h * C + c) * HU + k];
    feat2t[(h * 16 + k) * NN + n] = f;
    sa += f * ans[h * HU + k];
    na += f * ann[h * HU + k];
  }
#pragma unroll
  for (int k = HU; k < 16; ++k) feat2t[(h * 16 + k) * NN + n] = 0.f;
  s2[idx] = sa;
  n2[idx] = na;
}

// dense[h,i,m] = sum_j (s2[h,i] + n2[h,j]) * Einfo[j,m]
// A built in registers; B from transposed Einfo (Et[m][j]) -> float2 loads.
__global__ void dense_wmma_kernel(const float* __restrict__ s2,  // [8,128]
                                  const float* __restrict__ n2,  // [8,128]
                                  const float* __restrict__ Et,  // [128,128] = Einfo^T
                                  float* __restrict__ dense) {   // [8,128,128]
  int ntile = blockIdx.x, mtile = blockIdx.y, h = blockIdx.z;
  int lane = threadIdx.x & 31;
  int r = lane & 15;
  int kh = (lane >> 4) << 1;
  int hi = lane >> 4;
  float s2v = s2[h * NN + mtile * 16 + r];
  const float* n2h = n2 + h * NN + kh;
  const float* brow = Et + (ntile * 16 + r) * NN + kh;
  v8f c = {};
#pragma unroll 8
  for (int k0 = 0; k0 < NN; k0 += 4) {
    v2f nn2 = *(const v2f*)(n2h + k0);
    v2f a; a.x = s2v + nn2.x; a.y = s2v + nn2.y;
    v2f b = *(const v2f*)(brow + k0);
    c = wmma4(a, b, c);
  }
#pragma unroll
  for (int v = 0; v < 8; ++v) {
    int row = mtile * 16 + v + 8 * hi;
    dense[(h * NN + row) * NN + ntile * 16 + r] = c[v];
  }
}

// alpha[h,i,:] = softmax_m(leaky(dense[h,i,:]) + Aattr[i,:])
__global__ void alpha_softmax_kernel(const float* __restrict__ dense,
                                     const float* __restrict__ Aattr,  // [128,128]
                                     float* __restrict__ alpha) {
  int i = blockIdx.x, h = blockIdx.y, m = threadIdx.x;  // 128 threads
  __shared__ float red[128];
  float z = leaky02(dense[(h * NN + i) * NN + m]) + Aattr[i * NN + m];
  red[m] = z; __syncthreads();
  for (int s = 64; s > 0; s >>= 1) { if (m < s) red[m] = fmaxf(red[m], red[m + s]); __syncthreads(); }
  float mx = red[0]; __syncthreads();
  float w = expf(z - mx);
  red[m] = w; __syncthreads();
  for (int s = 64; s > 0; s >>= 1) { if (m < s) red[m] += red[m + s]; __syncthreads(); }
  alpha[(h * NN + i) * NN + m] = w / red[0];
}

// nf[h] = elu(alpha[h] @ feat2[h] + bn[h]); writes h1 layout and final output.
// B from feat2t[8][16][128] (padded) -> unconditional float2 loads.
template <int HU>
__global__ void nf_wmma_kernel(const float* __restrict__ alpha,   // [8,128,128]
                               const float* __restrict__ feat2t,  // [8,16,128]
                               const float* __restrict__ bn,      // [8,HU]
                               float* __restrict__ h1,            // [128, 8*HU]
                               float* __restrict__ out, int attr, int off) {
  int mtile = blockIdx.x, h = blockIdx.y;
  int lane = threadIdx.x & 31;
  int r = lane & 15;
  int kh = (lane >> 4) << 1;
  int hi = lane >> 4;
  const float* arow = alpha + h * NN * NN + (mtile * 16 + r) * NN + kh;
  const float* brow = feat2t + (h * 16 + r) * NN + kh;
  v8f c = {};
#pragma unroll 8
  for (int k0 = 0; k0 < NN; k0 += 4) {
    v2f a = *(const v2f*)(arow + k0);
    v2f b = *(const v2f*)(brow + k0);
    c = wmma4(a, b, c);
  }
  if (r < HU) {
    float bv = bn[h * HU + r];
#pragma unroll
    for (int v = 0; v < 8; ++v) {
      int row = mtile * 16 + v + 8 * hi;
      float x = c[v] + bv;
      x = x > 0.f ? x : expm1f(x);  // ELU
      h1[row * (HH * HU) + h * HU + r] = x;
      out[row * OUTW + attr * ATTRW + off + h * HU + r] = x;
    }
  }
}

// out[n, attr, 0:128] = X[n, :]
__global__ void copy_x_kernel(const float* __restrict__ X, float* __restrict__ out) {
  int idx = blockIdx.x * 256 + threadIdx.x;
  if (idx >= NN * FF) return;
  int n = idx / FF, f = idx % FF;
  float v = X[idx];
  for (int a = 0; a < 3; ++a) out[n * OUTW + a * ATTRW + f] = v;
}

// ---------------------------------------------------------------------------
extern "C" void kernel_launch(void* const* d_in, const int* in_sizes, int n_in,
                              void* d_out, int out_size, void* d_ws, size_t ws_size,
                              hipStream_t stream) {
  (void)in_sizes; (void)n_in; (void)out_size; (void)ws_size;
  const float* X    = (const float*)d_in[0];
  const float* Ab   = (const float*)d_in[1];   // A [3,128,128]
  const float* E    = (const float*)d_in[2];   // [3,128,128]
  const float* e2n  = (const float*)d_in[3];   // [128,16384]
  const float* Wn0  = (const float*)d_in[4];
  const float* bn0  = (const float*)d_in[5];
  const float* an0s = (const float*)d_in[6];
  const float* an0n = (const float*)d_in[7];
  const float* Wn1  = (const float*)d_in[8];
  const float* bn1  = (const float*)d_in[9];
  const float* an1s = (const float*)d_in[10];
  const float* an1n = (const float*)d_in[11];
  const float* We0  = (const float*)d_in[12];
  const float* be0  = (const float*)d_in[13];
  const float* We1  = (const float*)d_in[14];
  const float* be1  = (const float*)d_in[15];
  const float* Wct0 = (const float*)d_in[16];
  const float* bct0 = (const float*)d_in[17];
  const float* Wct1 = (const float*)d_in[18];
  const float* bct1 = (const float*)d_in[19];
  const float* ae0s = (const float*)d_in[20];
  const float* ae0n = (const float*)d_in[21];
  const float* ae1s = (const float*)d_in[22];
  const float* ae1n = (const float*)d_in[23];
  float* out = (float*)d_out;

  float* ws = (float*)d_ws;
  float* featE0 = ws; ws += MM * 32;       // [M,32]  (also layer-1 X_edge)
  float* featE1 = ws; ws += MM * 16;       // [M,16]
  float* aedge0 = ws; ws += HH * MM;
  float* aedge1 = ws; ws += HH * MM;
  float* feat   = ws; ws += HH * NN * 16;  // padded stride 16
  float* a_self = ws; ws += HH * NN;
  float* nfe    = ws; ws += HH * NN * 4;
  float* feat2t = ws; ws += HH * 16 * NN;  // transposed + padded
  float* s2     = ws; ws += HH * NN;
  float* n2     = ws; ws += HH * NN;
  float* dense  = ws; ws += HH * NN * NN;
  float* alpha  = ws; ws += HH * NN * NN;
  float* h1a    = ws; ws += NN * 128;      // layer-0 concat-heads output
  float* h1b    = ws; ws += NN * 64;       // layer-1 (scratch)
  float* Wt     = ws; ws += HH * 16 * FF;  // padded/transposed node weights
  float* Et     = ws; ws += NN * NN;       // transposed E_info

  copy_x_kernel<<<(NN * FF + 255) / 256, 256, 0, stream>>>(X, out);
  // Edge features are attr-independent: compute once.
  edge_feat0_kernel<<<MM / 256, 256, 0, stream>>>(E, We0, ae0n, featE0, aedge0);
  edge_feat1_kernel<<<MM / 256, 256, 0, stream>>>(featE0, We1, ae1n, featE1, aedge1);

  for (int attr = 0; attr < 3; ++attr) {
    const float* Aat = Ab + attr * NN * NN;
    // ---- layer 0 (HU=16, KE=4), E_info = E[attr] ----
    padt_weight_kernel<<<64, 256, 0, stream>>>(Wn0 + attr * HH * FF * 16, 16, Wt);
    node_feat_wmma_kernel<16><<<dim3(8, 8), 32, 0, stream>>>(X, Wt, feat);
    a_self_kernel<16><<<4, 256, 0, stream>>>(feat, ae0s, a_self);
    alpha_e_nfe_kernel<4, 32><<<dim3(NN, HH), 256, 0, stream>>>(a_self, aedge0, e2n,
                                                                featE0, be0, nfe);
    feat2_kernel<16, 4><<<4, 256, 0, stream>>>(feat, nfe, Wct0, bct0,
                                               an0s + attr * 128, an0n + attr * 128,
                                               feat2t, s2, n2);
    transpose128_kernel<<<64, 256, 0, stream>>>(E + attr * NN * NN, Et);
    dense_wmma_kernel<<<dim3(8, 8, 8), 32, 0, stream>>>(s2, n2, Et, dense);
    alpha_softmax_kernel<<<dim3(NN, HH), 128, 0, stream>>>(dense, Aat, alpha);
    nf_wmma_kernel<16><<<dim3(8, 8), 32, 0, stream>>>(alpha, feat2t, bn0 + attr * 128,
                                                      h1a, out, attr, 128);
    // ---- layer 1 (HU=8, KE=2), E_info = alpha[-1] from layer 0 ----
    padt_weight_kernel<<<64, 256, 0, stream>>>(Wn1 + attr * HH * FF * 8, 8, Wt);
    node_feat_wmma_kernel<8><<<dim3(8, 8), 32, 0, stream>>>(h1a, Wt, feat);
    a_self_kernel<8><<<4, 256, 0, stream>>>(feat, ae1s, a_self);
    alpha_e_nfe_kernel<2, 16><<<dim3(NN, HH), 256, 0, stream>>>(a_self, aedge1, e2n,
                                                                featE1, be1, nfe);
    feat2_kernel<8, 2><<<4, 256, 0, stream>>>(feat, nfe, Wct1, bct1,
                                              an1s + attr * 64, an1n + attr * 64,
                                              feat2t, s2, n2);
    transpose128_kernel<<<64, 256, 0, stream>>>(alpha + 7 * NN * NN, Et);  // E_info = alpha[-1]
    dense_wmma_kernel<<<dim3(8, 8, 8), 32, 0, stream>>>(s2, n2, Et, dense);
    alpha_softmax_kernel<<<dim3(NN, HH), 128, 0, stream>>>(dense, Aat, alpha);
    nf_wmma_kernel<8><<<dim3(8, 8), 32, 0, stream>>>(alpha, feat2t, bn1 + attr * 64,
                                                     h1b, out, attr, 256);
  }
}